// TopKSAE_3642132267024
// MI455X (gfx1250) — compile-verified
//
#include <hip/hip_runtime.h>
#include <hip/hip_bf16.h>
#include <float.h>

typedef __attribute__((ext_vector_type(2))) float v2f;
typedef __attribute__((ext_vector_type(8))) float v8f;

typedef unsigned __attribute__((ext_vector_type(4))) tdm_g0_t;  // D# group0
typedef int      __attribute__((ext_vector_type(8))) tdm_g1_t;  // D# group1
typedef int      __attribute__((ext_vector_type(4))) tdm_g23_t; // D# group2/3

constexpr int B = 8192, D = 512, F = 32768, K = 24;
constexpr int MT = 16;          // rows per block
constexpr int NT = 128;         // features per outer iteration (4 waves x 2 tiles x 16)
constexpr int KB = 32;          // k-slice per TDM transfer
constexpr int NTHREADS = 128;   // 4 waves (wave32)
constexpr int AS_STRIDE = 516;  // (4*m + k) % 64 distinct for m=0..15 -> conflict-free b64 reads
constexpr int BS_STRIDE = 36;   // 32 data + 4 pad dwords (TDM pad_interval=32dw, pad_amount=4dw)

// ---------------------------------------------------------------------------
// TDM: DMA one [NT x KB] fp32 weight slice (row stride D) into LDS with a
// 4-dword pad every 32 dwords -> LDS row stride = BS_STRIDE floats.
// Descriptor layout per CDNA5 ISA.
// ---------------------------------------------------------------------------
__device__ __forceinline__ void tdm_load_b_slice(const float* gsrc,
                                                 unsigned lds_byte_off) {
  unsigned long long ga = (unsigned long long)(size_t)gsrc;
  tdm_g0_t g0;
  g0[0] = 1u;                                            // count=1, user mode
  g0[1] = lds_byte_off;                                  // lds_addr
  g0[2] = (unsigned)(ga & 0xFFFFFFFFull);                // global_addr[31:0]
  g0[3] = (unsigned)((ga >> 32) & 0x01FFFFFFull)         // global_addr[56:32]
        | (2u << 30);                                    // type=2 ("image")
  tdm_g1_t g1;
  g1[0] = (2 << 16)                                      // data_size = 4B
        | (1 << 20)                                      // pad_enable
        | (4 << 22)                                      // pad_interval: 32 dwords
        | (3 << 25);                                     // pad_amount: 4 dwords
  g1[1] = (int)((unsigned)KB << 16);                     // tensor_dim0[15:0] = 32
  g1[2] = (int)((unsigned)NT << 16);                     // tensor_dim1[15:0] = 128
  g1[3] = (int)((unsigned)KB << 16);                     // tile_dim0 = 32
  g1[4] = NT;                                            // tile_dim1 = 128
  g1[5] = D;                                             // tensor_dim0_stride = 512
  g1[6] = 0;
  g1[7] = 0;
  tdm_g23_t gz = {0, 0, 0, 0};
#if __clang_major__ >= 23
  tdm_g1_t gz8 = {0, 0, 0, 0, 0, 0, 0, 0};
  __builtin_amdgcn_tensor_load_to_lds(g0, g1, gz, gz, gz8, 0);
#else
  __builtin_amdgcn_tensor_load_to_lds(g0, g1, gz, gz, 0);
#endif
}

// ---------------------------------------------------------------------------
// Kernel 1: enc = (x - bias) @ enc_w^T + enc_b  via V_WMMA_F32_16X16X4_F32.
// B slices streamed by the Tensor Data Mover (double-buffered, TENSORcnt).
// Each wave register-blocks two 16x16 N-tiles (independent chains c0/c1).
// ---------------------------------------------------------------------------
__global__ __launch_bounds__(NTHREADS) void sae_encode_gemm(
    const float* __restrict__ x, const float* __restrict__ enc_w,
    const float* __restrict__ enc_b, const float* __restrict__ bias,
    float* __restrict__ enc_out)
{
  __shared__ float As[MT * AS_STRIDE];
  __shared__ float Bs[2][NT * BS_STRIDE];

  const int tid    = threadIdx.x;
  const int lane   = tid & 31;
  const int wave   = tid >> 5;        // 0..3
  const int n_lane = lane & 15;
  const int khalf  = lane >> 4;       // 0 or 1 (wave32 halves)
  const int row0   = blockIdx.x * MT;

  const unsigned bs_off0 = (unsigned)(size_t)&Bs[0][0];  // LDS byte offsets
  const unsigned bs_off1 = (unsigned)(size_t)&Bs[1][0];

  // Stage A = (x - bias) tile [16 x 512], coalesced float4 loads.
  #pragma unroll
  for (int it = 0; it < (MT * D / 4) / NTHREADS; ++it) {
    int idx = tid + NTHREADS * it;            // 0..2047 float4 slots
    int m   = idx >> 7;                       // / (D/4)
    int q   = idx & 127;
    const float4 xv = ((const float4*)(x + (size_t)(row0 + m) * D))[q];
    const float4 bv = ((const float4*)bias)[q];
    float* dst = &As[m * AS_STRIDE + q * 4];
    dst[0] = xv.x - bv.x; dst[1] = xv.y - bv.y;
    dst[2] = xv.z - bv.z; dst[3] = xv.w - bv.w;
  }
  __syncthreads();

  for (int f0 = 0; f0 < F; f0 += NT) {
    // Warm L2 for the next feature chunk while we compute this one.
    if (f0 + NT < F)
      __builtin_prefetch(enc_w + (size_t)(f0 + NT) * D + (size_t)tid * 32, 0, 1);

    // Prologue: DMA first k-slice into buffer 0.
    if (wave == 0) {
      tdm_load_b_slice(enc_w + (size_t)f0 * D, bs_off0);
      __builtin_amdgcn_s_wait_tensorcnt(0);
    }
    __syncthreads();

    v8f c0 = {};                              // accumulators: tile n = wave*2
    v8f c1 = {};                              // accumulators: tile n = wave*2+1
    for (int kb = 0; kb < D; kb += KB) {
      const int cur = (kb / KB) & 1;
      // Issue DMA for the next k-slice into the other buffer (no wait yet).
      if (wave == 0 && (kb + KB) < D)
        tdm_load_b_slice(enc_w + (size_t)f0 * D + (kb + KB),
                         cur ? bs_off0 : bs_off1);

      const float* ap  = &As[n_lane * AS_STRIDE + kb + 2 * khalf];
      const float* bp0 = &Bs[cur][(wave * 32 + n_lane) * BS_STRIDE + 2 * khalf];
      const float* bp1 = bp0 + 16 * BS_STRIDE;
      #pragma unroll
      for (int k = 0; k < KB; k += 4) {
        v2f a  = *(const v2f*)(ap + k);       // A: M=n_lane, K={k+2*khalf, +1}
        v2f b0 = *(const v2f*)(bp0 + k);      // B tile 0: N=n_lane
        v2f b1 = *(const v2f*)(bp1 + k);      // B tile 1: N=n_lane+16
        c0 = __builtin_amdgcn_wmma_f32_16x16x4_f32(
            false, a, false, b0, (short)0, c0, false, false);
        c1 = __builtin_amdgcn_wmma_f32_16x16x4_f32(
            false, a, false, b1, (short)0, c1, false, false);
      }

      if (wave == 0 && (kb + KB) < D)
        __builtin_amdgcn_s_wait_tensorcnt(0); // next buffer landed
      __syncthreads();                         // everyone done with 'cur'
    }

    // Epilogue: add enc_b, store tiles (C layout: M = i + 8*khalf, N = n_lane).
    const int   colg = f0 + wave * 32 + n_lane;
    const float eb0  = enc_b[colg];
    const float eb1  = enc_b[colg + 16];
    #pragma unroll
    for (int i = 0; i < 8; ++i) {
      int r = row0 + i + 8 * khalf;
      enc_out[(size_t)r * F + colg]      = c0[i] + eb0;
      enc_out[(size_t)r * F + colg + 16] = c1[i] + eb1;
    }
  }
}

// ---------------------------------------------------------------------------
// Kernel 2: exact per-row top-K, write values/indices, mask row in place.
// ---------------------------------------------------------------------------
__global__ __launch_bounds__(1024) void sae_topk(
    float* __restrict__ enc_inout, float* __restrict__ out_idx,
    float* __restrict__ out_val)
{
  constexpr int J = F / 1024;                 // 32 values per thread, in regs
  __shared__ float sval[1024];
  __shared__ int   sidx[1024];

  const int tid = threadIdx.x;
  const int b   = blockIdx.x;
  float* row = enc_inout + (size_t)b * F;

  float vals[J], work[J];
  #pragma unroll
  for (int j = 0; j < J; ++j) { vals[j] = row[tid + 1024 * j]; work[j] = vals[j]; }

  unsigned sel = 0u;
  for (int it = 0; it < K; ++it) {
    float lv = -FLT_MAX; int lj = 0;
    #pragma unroll
    for (int j = 0; j < J; ++j)
      if (work[j] > lv) { lv = work[j]; lj = j; }
    sval[tid] = lv;
    sidx[tid] = tid + 1024 * lj;              // == feature index
    __syncthreads();
    for (int s = 512; s > 0; s >>= 1) {
      if (tid < s) {
        float v2 = sval[tid + s]; int i2 = sidx[tid + s];
        if (v2 > sval[tid] || (v2 == sval[tid] && i2 < sidx[tid])) {
          sval[tid] = v2; sidx[tid] = i2;
        }
      }
      __syncthreads();
    }
    const int   fidx = sidx[0];
    const float wv   = sval[0];
    __syncthreads();                          // sval/sidx reused next iteration
    if ((fidx & 1023) == tid) {
      work[fidx >> 10] = -FLT_MAX;
      sel |= 1u << (fidx >> 10);
    }
    if (tid == it) {
      out_val[(size_t)b * K + it] = wv;
      out_idx[(size_t)b * K + it] = (float)fidx;   // fp32 output buffer
    }
  }
  // Masked "encoded" output: keep only top-K entries.
  #pragma unroll
  for (int j = 0; j < J; ++j)
    row[tid + 1024 * j] = ((sel >> j) & 1u) ? vals[j] : 0.0f;
}

// ---------------------------------------------------------------------------
// Kernel 3: decoded = encoded @ dec_w^T + bias (sparse, K=24); MSE partials.
// ---------------------------------------------------------------------------
__global__ __launch_bounds__(256) void sae_decode(
    const float* __restrict__ x, const float* __restrict__ dec_w,
    const float* __restrict__ bias, const float* __restrict__ out_idx,
    const float* __restrict__ out_val, float* __restrict__ decoded,
    float* __restrict__ partial)
{
  __shared__ float sred[256];
  __shared__ float sv[K];
  __shared__ int   si[K];
  const int b = blockIdx.x, tid = threadIdx.x;
  if (tid < K) {
    sv[tid] = out_val[(size_t)b * K + tid];
    si[tid] = (int)out_idx[(size_t)b * K + tid];
  }
  __syncthreads();

  float errsum = 0.f;
  for (int d = tid; d < D; d += 256) {
    float acc = bias[d];
    #pragma unroll
    for (int k = 0; k < K; ++k)
      acc += sv[k] * dec_w[(size_t)d * F + si[k]];    // dec_w L2-resident
    decoded[(size_t)b * D + d] = acc;
    float e = (x[(size_t)b * D + d] - bias[d]) - acc;
    errsum += e * e;
  }
  sred[tid] = errsum;
  __syncthreads();
  for (int s = 128; s > 0; s >>= 1) {
    if (tid < s) sred[tid] += sred[tid + s];
    __syncthreads();
  }
  if (tid == 0) partial[b] = sred[0];
}

// ---------------------------------------------------------------------------
// Kernel 4: deterministic final MSE reduction.
// ---------------------------------------------------------------------------
__global__ __launch_bounds__(256) void sae_mse(
    const float* __restrict__ partial, float* __restrict__ mse)
{
  __shared__ float sred[256];
  float s = 0.f;
  for (int i = threadIdx.x; i < B; i += 256) s += partial[i];
  sred[threadIdx.x] = s;
  __syncthreads();
  for (int st = 128; st > 0; st >>= 1) {
    if (threadIdx.x < st) sred[threadIdx.x] += sred[threadIdx.x + st];
    __syncthreads();
  }
  if (threadIdx.x == 0) *mse = sred[0] / (float)((size_t)B * D);
}

// ---------------------------------------------------------------------------
extern "C" void kernel_launch(void* const* d_in, const int* in_sizes, int n_in,
                              void* d_out, int out_size, void* d_ws, size_t ws_size,
                              hipStream_t stream)
{
  (void)in_sizes; (void)n_in; (void)out_size; (void)ws_size;
  const float* x     = (const float*)d_in[0];
  const float* enc_w = (const float*)d_in[1];
  const float* enc_b = (const float*)d_in[2];
  const float* dec_w = (const float*)d_in[3];
  const float* bias  = (const float*)d_in[4];
  // d_in[5] = topk (== 24, baked in as compile-time K)

  float* out      = (float*)d_out;
  float* encoded  = out;                               // [B*F]
  float* decoded  = encoded + (size_t)B * F;           // [B*D]
  float* mse      = decoded + (size_t)B * D;           // [1]
  float* indices  = mse + 1;                           // [B*K] (as float)
  float* values   = indices + (size_t)B * K;           // [B*K]
  float* partial  = (float*)d_ws;                      // [B] scratch

  sae_encode_gemm<<<B / MT, NTHREADS, 0, stream>>>(x, enc_w, enc_b, bias, encoded);
  sae_topk      <<<B, 1024, 0, stream>>>(encoded, indices, values);
  sae_decode    <<<B, 256, 0, stream>>>(x, dec_w, bias, indices, values, decoded, partial);
  sae_mse       <<<1, 256, 0, stream>>>(partial, mse);
}